// RayObsGraph_25769804174
// MI455X (gfx1250) — compile-verified
//
#include <hip/hip_runtime.h>

typedef __attribute__((ext_vector_type(2))) float v2f;
typedef __attribute__((ext_vector_type(4))) float v4f;
typedef __attribute__((ext_vector_type(8))) float v8f;
typedef __attribute__((ext_vector_type(4))) int v4i;

#if defined(__has_builtin)
#if __has_builtin(__builtin_amdgcn_global_load_async_to_lds_b128) && \
    __has_builtin(__builtin_amdgcn_s_wait_asynccnt)
#define HAVE_ASYNC_LDS 1
#endif
#endif
#ifndef HAVE_ASYNC_LDS
#define HAVE_ASYNC_LDS 0
#endif

namespace {
constexpr int kB = 32, kT = 32, kN = 256, kOBS = 512, kH = 256, kOUT = 256, kA = 18;
constexpr int KSTEP = 32;    // K tile per LDS panel
constexpr int ROWW = 80;     // padded panel row stride (words): conflict-free 2addr reads
}

#if HAVE_ASYNC_LDS
typedef __attribute__((address_space(1))) v4i* as1_v4i_ptr;
typedef __attribute__((address_space(3))) v4i* as3_v4i_ptr;

__device__ __forceinline__ void async_copy16(const float* g, float* l) {
  // GLOBAL_LOAD_ASYNC_TO_LDS_B128: DMA 16 bytes global -> LDS, tracked by ASYNCcnt.
  __builtin_amdgcn_global_load_async_to_lds_b128(
      (as1_v4i_ptr)(void*)g,   // drop const, then generic->AS1 (flat global == AS1)
      (as3_v4i_ptr)l,          // addrspacecast generic->AS3 (strips LDS aperture)
      0, 0);
}
#endif

// dinv[b,i] = rsqrt(max(1, 1 + sum_j adj[b,i,j]))
__global__ __launch_bounds__(256) void degree_kernel(const float* __restrict__ adj,
                                                     float* __restrict__ dinv) {
  int idx = blockIdx.x * blockDim.x + threadIdx.x;  // b*N + i
  if (idx >= kB * kN) return;
  const float* row = adj + (long long)idx * kN;
  float s = 1.0f;  // self loop
  for (int j = 0; j < kN; j += 4)
    s += row[j] + row[j + 1] + row[j + 2] + row[j + 3];
  s = fmaxf(s, 1.0f);
  dinv[idx] = rsqrtf(s);
}

// An[b,i,j] = dinv[b,i] * (adj[b,i,j] + (i==j)) * dinv[b,j]
__global__ __launch_bounds__(256) void norm_adj_kernel(const float* __restrict__ adj,
                                                       const float* __restrict__ dinv,
                                                       float* __restrict__ An) {
  long long idx = (long long)blockIdx.x * blockDim.x + threadIdx.x;
  if (idx >= (long long)kB * kN * kN) return;
  int j = (int)(idx % kN);
  long long bi = idx / kN;
  int i = (int)(bi % kN);
  int b = (int)(bi / kN);
  float a = adj[idx] + (i == j ? 1.0f : 0.0f);
  An[idx] = dinv[b * kN + i] * a * dinv[b * kN + j];
}

// nodes_f = nodes0 with x_t scattered to slot min(nn0+t, N-1) for t in [0,T); last write wins.
__global__ __launch_bounds__(256) void build_nodes_kernel(const float* __restrict__ flat,
                                                          const float* __restrict__ nodes0,
                                                          const int* __restrict__ nn0,
                                                          float* __restrict__ nodesF) {
  long long idx = (long long)blockIdx.x * blockDim.x + threadIdx.x;
  if (idx >= (long long)kB * kN * kOBS) return;
  int o = (int)(idx % kOBS);
  long long bi = idx / kOBS;
  int i = (int)(bi % kN);
  int b = (int)(bi / kN);
  int n0 = nn0[b];
  int t = -1;
  if (i == kN - 1) {
    if (n0 + (kT - 1) >= kN - 1) t = kT - 1;  // clamped slot: last write wins
  } else {
    int r = i - n0;
    if (r >= 0 && r < kT) t = r;              // unique writer
  }
  float v = (t >= 0) ? flat[((long long)b * kT + t) * kOBS + o] : nodes0[idx];
  nodesF[idx] = v;
}

// Batched C[b](MxNC) = act(A[b](MxK) @ Bm[b](KxNC) + bias).  strideB==0 -> shared weights.
// Block: 8 waves -> 128x64 tile. B panel (KSTEP x 64) double-buffered in LDS.
// Panel movement: async global->LDS DMA (ASYNCcnt) when available, else regs + ds_store.
__global__ __launch_bounds__(256) void gemm_wmma_f32(const float* __restrict__ A,
                                                     const float* __restrict__ Bm,
                                                     float* __restrict__ C,
                                                     const float* __restrict__ bias,
                                                     int M, int K, int NC,
                                                     long long strideA, long long strideB,
                                                     long long strideC, int relu) {
  __shared__ float ldsB[2][KSTEP][ROWW];  // row-major panel, padded; 20KB

  const int tid = threadIdx.x;
  const int lane = tid & 31;
  const int wave = tid >> 5;                 // 0..7 -> 16-row subtile
  const int colBlocks = NC >> 6;             // 64-col blocks
  const int rowBlocks = M >> 7;              // 128-row blocks
  const int blocksPerBatch = rowBlocks * colBlocks;
  int gb = blockIdx.x;
  int b = gb / blocksPerBatch;
  int r = gb % blocksPerBatch;
  int bm = (r / colBlocks) << 7;
  int bn = (r % colBlocks) << 6;

  const float* Ab = A + (long long)b * strideA;
  const float* Bb = Bm + (long long)b * strideB + bn;
  float* Cb = C + (long long)b * strideC;

  const int nl = lane & 15;  // M index (A) / N index (B,C)
  const int kh = lane >> 4;  // K-half selector
  const int tm = bm + (wave << 4);
  const float* Arow = Ab + (long long)(tm + nl) * K + 2 * kh;

  // B panel loader: pair p = tid>>4 (rows 2p, 2p+1), quad q = tid&15 (cols 4q..4q+3)
  const int lp2 = (tid >> 4) << 1;   // even panel row
  const int lq = (tid & 15) << 2;
  const float* Bld = Bb + (long long)lp2 * NC + lq;

#if HAVE_ASYNC_LDS
  // issue panel 0 DMA
  async_copy16(Bld, &ldsB[0][lp2][lq]);
  async_copy16(Bld + NC, &ldsB[0][lp2 + 1][lq]);
#else
  // preload panel 0 into registers
  v4f pr0 = *(const v4f*)(Bld);
  v4f pr1 = *(const v4f*)(Bld + NC);
#endif

  v8f acc[4] = {};
  const int steps = K / KSTEP;
  for (int s = 0; s < steps; ++s) {
#if !HAVE_ASYNC_LDS
    // commit current panel regs -> LDS (row-major b128 stores)
    *(v4f*)&ldsB[s & 1][lp2][lq] = pr0;
    *(v4f*)&ldsB[s & 1][lp2 + 1][lq] = pr1;
#endif
    // A fragments for this K-step (independent global clause; latency overlaps barrier)
    v2f afr[KSTEP / 4];
#pragma unroll
    for (int i = 0; i < KSTEP / 4; ++i)
      afr[i] = *(const v2f*)(Arow + s * KSTEP + 4 * i);
#if HAVE_ASYNC_LDS
    __builtin_amdgcn_s_wait_asynccnt(0);  // this wave's panel-s DMA complete
    __syncthreads();                      // all waves' DMA complete -> panel s ready
    if (s + 1 < steps) {                  // safe: prior readers of buf (s+1)&1 passed barrier
      const float* nb = Bld + (long long)(s + 1) * KSTEP * NC;
      async_copy16(nb, &ldsB[(s + 1) & 1][lp2][lq]);
      async_copy16(nb + NC, &ldsB[(s + 1) & 1][lp2 + 1][lq]);
    }
#else
    // prefetch next B panel into registers
    if (s + 1 < steps) {
      const float* nb = Bld + (long long)(s + 1) * KSTEP * NC;
      pr0 = *(const v4f*)(nb);
      pr1 = *(const v4f*)(nb + NC);
    }
    __syncthreads();  // panel visible; prior-buffer readers finished one barrier ago
#endif
    // 32 WMMAs fed from LDS (fragment = rows k,k+1 at column nl+16c -> ds_load_2addr_b32)
#pragma unroll
    for (int i = 0; i < KSTEP / 4; ++i) {
      const int k = 4 * i + 2 * kh;
#pragma unroll
      for (int c = 0; c < 4; ++c) {
        v2f bbv;
        bbv.x = ldsB[s & 1][k][nl + 16 * c];
        bbv.y = ldsB[s & 1][k + 1][nl + 16 * c];
        acc[c] = __builtin_amdgcn_wmma_f32_16x16x4_f32(false, afr[i], false, bbv,
                                                       (short)0, acc[c], false, false);
      }
    }
  }

#pragma unroll
  for (int c = 0; c < 4; ++c) {
    int n = bn + 16 * c + nl;
    float bz = bias ? bias[n] : 0.0f;
#pragma unroll
    for (int v = 0; v < 8; ++v) {
      float val = acc[c][v] + bz;
      if (relu) val = fmaxf(val, 0.0f);
      Cb[(long long)(tm + v + 8 * kh) * NC + n] = val;
    }
  }
}

__global__ __launch_bounds__(256) void zero_kernel(float* __restrict__ p, int n) {
  int i = blockIdx.x * blockDim.x + threadIdx.x;
  if (i < n) p[i] = 0.0f;
}

// out_last[b] = h3[b, min(nn0+T-1, N-1)]; logits[b,T-1] = out_last@Wl + bl; value likewise.
__global__ void head_kernel(const float* __restrict__ h3, const int* __restrict__ nn0,
                            const float* __restrict__ Wl, const float* __restrict__ bl,
                            const float* __restrict__ Wv, const float* __restrict__ bv,
                            float* __restrict__ logits, float* __restrict__ values,
                            int* __restrict__ nnF) {
  int b = blockIdx.x;
  int lane = threadIdx.x;
  int n0 = nn0[b];
  int nlast = min(n0 + kT - 1, kN - 1);
  const float* o = h3 + ((long long)b * kN + nlast) * kOUT;
  if (lane < kA) {
    float s = bl[lane];
    for (int k = 0; k < kOUT; ++k) s += o[k] * Wl[k * kA + lane];
    logits[((long long)b * kT + (kT - 1)) * kA + lane] = s;
  } else if (lane == kA) {
    float s = bv[0];
    for (int k = 0; k < kOUT; ++k) s += o[k] * Wv[k];
    values[b * kT + (kT - 1)] = s;
  } else if (lane == kA + 1) {
    nnF[b] = min(n0 + kT, kN - 1);
  }
}

extern "C" void kernel_launch(void* const* d_in, const int* in_sizes, int n_in,
                              void* d_out, int out_size, void* d_ws, size_t ws_size,
                              hipStream_t stream) {
  const float* flat   = (const float*)d_in[0];
  const float* nodes0 = (const float*)d_in[1];
  const float* adj0   = (const float*)d_in[2];
  const float* W0 = (const float*)d_in[3];
  const float* b0 = (const float*)d_in[4];
  const float* W1 = (const float*)d_in[5];
  const float* b1 = (const float*)d_in[6];
  const float* W2 = (const float*)d_in[7];
  const float* b2 = (const float*)d_in[8];
  const float* Wl = (const float*)d_in[9];
  const float* bl = (const float*)d_in[10];
  const float* Wv = (const float*)d_in[11];
  const float* bv = (const float*)d_in[12];
  const int*   nn0 = (const int*)d_in[13];
  (void)in_sizes; (void)n_in; (void)out_size; (void)ws_size;

  // d_out layout: logits | values | nodes_f | nn_f
  float* out    = (float*)d_out;
  float* logits = out;                                         // B*T*A
  float* values = logits + kB * kT * kA;                       // B*T
  float* nodesF = values + kB * kT;                            // B*N*OBS
  int*   nnF    = (int*)(nodesF + (long long)kB * kN * kOBS);  // B (int bits)

  // workspace: dinv | An | buf0 | buf1
  float* ws   = (float*)d_ws;
  float* dinv = ws;                                            // B*N
  float* An   = dinv + kB * kN;                                // B*N*N
  float* buf0 = An + (long long)kB * kN * kN;                  // B*N*H
  float* buf1 = buf0 + (long long)kB * kN * kH;                // B*N*H

  const long long sNOBS = (long long)kN * kOBS;
  const long long sNN   = (long long)kN * kN;
  const long long sNH   = (long long)kN * kH;

  degree_kernel<<<(kB * kN + 255) / 256, 256, 0, stream>>>(adj0, dinv);
  norm_adj_kernel<<<(int)(((long long)kB * kN * kN + 255) / 256), 256, 0, stream>>>(adj0, dinv, An);
  build_nodes_kernel<<<(int)(((long long)kB * kN * kOBS + 255) / 256), 256, 0, stream>>>(
      flat, nodes0, nn0, nodesF);

  // GEMM grid: B * (M/128) * (NC/64) blocks of 256 threads
  const int gemmBlocks = kB * (kN / 128) * (kH / 64);  // 256

  // Y0 = nodes_f @ W0
  gemm_wmma_f32<<<gemmBlocks, 256, 0, stream>>>(nodesF, W0, buf0, nullptr,
                                                kN, kOBS, kH, sNOBS, 0, sNH, 0);
  // h1 = relu(An @ Y0 + b0)
  gemm_wmma_f32<<<gemmBlocks, 256, 0, stream>>>(An, buf0, buf1, b0,
                                                kN, kN, kH, sNN, sNH, sNH, 1);
  // Y1 = h1 @ W1
  gemm_wmma_f32<<<gemmBlocks, 256, 0, stream>>>(buf1, W1, buf0, nullptr,
                                                kN, kH, kH, sNH, 0, sNH, 0);
  // h2 = relu(An @ Y1 + b1)
  gemm_wmma_f32<<<gemmBlocks, 256, 0, stream>>>(An, buf0, buf1, b1,
                                                kN, kN, kH, sNN, sNH, sNH, 1);
  // Y2 = h2 @ W2
  gemm_wmma_f32<<<gemmBlocks, 256, 0, stream>>>(buf1, W2, buf0, nullptr,
                                                kN, kH, kOUT, sNH, 0, sNH, 0);
  // h3 = An @ Y2 + b2
  gemm_wmma_f32<<<gemmBlocks, 256, 0, stream>>>(An, buf0, buf1, b2,
                                                kN, kN, kOUT, sNN, sNH, sNH, 0);

  const int zn = kB * kT * kA + kB * kT;
  zero_kernel<<<(zn + 255) / 256, 256, 0, stream>>>(logits, zn);
  head_kernel<<<kB, 32, 0, stream>>>(buf1, nn0, Wl, bl, Wv, bv, logits, values, nnF);
}